// _global_attention_sop_u_22814866277110
// MI455X (gfx1250) — compile-verified
//
#include <hip/hip_runtime.h>

// CDNA5 / gfx1250: wave32, WMMA 16x16x4 f32.
typedef __attribute__((ext_vector_type(2))) float v2f;
typedef __attribute__((ext_vector_type(4))) float v4f;
typedef __attribute__((ext_vector_type(8))) float v8f;

#define DCOLS      256      // feature dim d (fixed by reference)
#define LSTRIDE    260      // padded LDS row stride in floats (4-float skew -> conflict-free strided B reads)
#define TILE_ROWS  32       // rows staged per tile
#define RPB        512      // rows per block (16 tiles)
#define NTHREADS   256      // 8 waves of 32

__global__ void init_out_kernel(float* __restrict__ out, const float* __restrict__ b, int n) {
  int i = blockIdx.x * blockDim.x + threadIdx.x;
  if (i < n) out[i] = b[0];
}

// GUARD=true only for the single partial block at the tail.
template <bool GUARD>
__device__ __forceinline__ void seg_sop_body(
    const float* __restrict__ x, const int* __restrict__ batch,
    const float* __restrict__ w, float* __restrict__ out,
    int M, float* ldsTile, float (*s_part)[TILE_ROWS])
{
  const int tid   = threadIdx.x;
  const int lane  = tid & 31;
  const int wave  = tid >> 5;            // 0..7
  const int sub   = wave & 1;            // 16-row subtile within the 32-row tile
  const int quar  = wave >> 1;           // K quarter: 0..3 (64 cols each)
  const int half  = lane >> 4;           // 0/1: which K-pair of the 4-wide chunk
  const int ncol  = lane & 15;           // N index (row within subtile) for B/C frags

  const int rowBase0 = blockIdx.x * RPB;

  float acc[8];
#pragma unroll
  for (int j = 0; j < 8; ++j) acc[j] = 0.0f;
  int curSeg = -1;

  const int nTiles = RPB / TILE_ROWS;    // 16
  for (int t = 0; t < nTiles; ++t) {
    const int rowBase = rowBase0 + t * TILE_ROWS;

    // ---------------- stage tile: global (coalesced, non-temporal b128) -> LDS --
    {
      const v4f* xs = (const v4f*)x;     // 64 float4 per row
#pragma unroll
      for (int i = tid; i < TILE_ROWS * 64; i += NTHREADS) {
        const int r  = i >> 6;
        const int c4 = i & 63;
        v4f v = {0.f, 0.f, 0.f, 0.f};
        if (!GUARD || (rowBase + r < M))
          v = __builtin_nontemporal_load(&xs[(long)(rowBase + r) * 64 + c4]);
        *(v4f*)&ldsTile[r * LSTRIDE + c4 * 4] = v;
      }
    }
    __syncthreads();

    // ---------------- phase A: per-row dot products via WMMA f32 16x16x4 -------
    // A[m][k] = w[k0+k] for all m (replicated) ; B[k][n] = x[row n, k0+k]
    // => C[m][n] = partial s_n in every row m; lane n holds s_n in c[0].
    {
      v8f cacc = {};
      const float* bbase = &ldsTile[(sub * 16 + ncol) * LSTRIDE];
#pragma unroll
      for (int c = 0; c < 16; ++c) {
        const int k0 = quar * 64 + c * 4 + 2 * half;
        v2f afrag = *(const v2f*)(w + k0);          // replicated w chunk (hoisted to regs)
        v2f bfrag = *(const v2f*)(bbase + k0);      // ds_load_b64, conflict-free
        cacc = __builtin_amdgcn_wmma_f32_16x16x4_f32(
            false, afrag, false, bfrag, (short)0, cacc, false, false);
      }
      if (lane < 16) s_part[quar][sub * 16 + ncol] = cacc[0];
    }
    __syncthreads();

    // ---------------- phase B: acc[g] += s_k * x_k (segment-local in regs) -----
#pragma unroll
    for (int i = 0; i < 4; ++i) {
      const int lr = wave * 4 + i;                  // tile-local row
      const int k  = rowBase + lr;                  // global row (wave-uniform)
      if (!GUARD || (k < M)) {
        // batch[k] is identical across the wave -> force scalar control flow
        const int g = __builtin_amdgcn_readfirstlane(batch[k]);
        const float s = s_part[0][lr] + s_part[1][lr] + s_part[2][lr] + s_part[3][lr];
        if (g != curSeg) {
          if (curSeg >= 0) {
            float* o = out + (long)curSeg * DCOLS + lane * 8;
#pragma unroll
            for (int j = 0; j < 8; ++j) unsafeAtomicAdd(o + j, acc[j]);
          }
          curSeg = g;
#pragma unroll
          for (int j = 0; j < 8; ++j) acc[j] = 0.0f;
        }
        const float* xr = &ldsTile[lr * LSTRIDE + lane * 8];
        const v4f xa = *(const v4f*)(xr);
        const v4f xb = *(const v4f*)(xr + 4);
#pragma unroll
        for (int j = 0; j < 4; ++j) acc[j]     = fmaf(s, xa[j], acc[j]);
#pragma unroll
        for (int j = 0; j < 4; ++j) acc[4 + j] = fmaf(s, xb[j], acc[4 + j]);
      }
    }
    __syncthreads();   // before next tile overwrites LDS
  }

  // final flush
  if (curSeg >= 0) {
    float* o = out + (long)curSeg * DCOLS + lane * 8;
#pragma unroll
    for (int j = 0; j < 8; ++j) unsafeAtomicAdd(o + j, acc[j]);
  }
}

__global__ __launch_bounds__(NTHREADS)
void seg_sop_kernel(const float* __restrict__ x,
                    const int*   __restrict__ batch,
                    const float* __restrict__ w,
                    float*       __restrict__ out,
                    int M)
{
  __shared__ float ldsTile[TILE_ROWS * LSTRIDE];   // ~33.3 KB
  __shared__ float s_part[4][TILE_ROWS];           // K-split partial dots

  if ((blockIdx.x + 1) * RPB <= (unsigned)M) {
    seg_sop_body<false>(x, batch, w, out, M, ldsTile, s_part);  // interior: no guards
  } else {
    seg_sop_body<true>(x, batch, w, out, M, ldsTile, s_part);   // tail block
  }
}

extern "C" void kernel_launch(void* const* d_in, const int* in_sizes, int n_in,
                              void* d_out, int out_size, void* d_ws, size_t ws_size,
                              hipStream_t stream) {
  const float* x     = (const float*)d_in[0];   // [M, 256] f32
  const int*   batch = (const int*)d_in[1];     // [M] sorted segment ids
  const float* w     = (const float*)d_in[2];   // [256] f32
  const float* b     = (const float*)d_in[3];   // [1] f32
  float* out = (float*)d_out;                   // [512, 256] f32

  const int M = in_sizes[1];

  // out = b
  init_out_kernel<<<(out_size + 255) / 256, 256, 0, stream>>>(out, b, out_size);

  // out[g] += sum_{k in g} (x_k . w) * x_k
  const int nb = (M + RPB - 1) / RPB;
  seg_sop_kernel<<<nb, NTHREADS, 0, stream>>>(x, batch, w, out, M);
}